// FasterRCNN_23235773071452
// MI455X (gfx1250) — compile-verified
//
#include <hip/hip_runtime.h>
#include <hip/hip_bf16.h>

// ---------------- problem constants ----------------
#define H_FEAT 512
#define W_FEAT 512
#define K_ANCH 9
#define N_ANCH (H_FEAT * W_FEAT * K_ANCH)   // 2,359,296
#define FIELD  32.0f
#define IMG_W  16384.0f
#define IMG_H  16384.0f
#define TOPK   4000
#define WORDS  125                           // 4000 / 32
#define CANDMAX 4096
#define THRESH 0.5f
#define NMS_IOU 0.3f

typedef __attribute__((ext_vector_type(2))) float v2f;
typedef __attribute__((ext_vector_type(8))) float v8f;

#if __has_builtin(__builtin_amdgcn_global_load_async_to_lds_b32) && \
    __has_builtin(__builtin_amdgcn_s_wait_asynccnt)
#define HAVE_ASYNC_LDS 1
typedef __attribute__((address_space(1))) int GlobI32;   // builtin expects int* (AS1)
typedef __attribute__((address_space(3))) int LdsI32;    // builtin expects int* (AS3)
#endif

// sortable key: monotonic float->uint (descending top-k = largest keys)
__device__ __forceinline__ unsigned f2key(float f) {
    unsigned u = __float_as_uint(f);
    return (u & 0x80000000u) ? ~u : (u | 0x80000000u);
}
__device__ __forceinline__ float key2f(unsigned s) {
    unsigned u = (s & 0x80000000u) ? (s & 0x7FFFFFFFu) : ~s;
    return __uint_as_float(u);
}

// ---------------- stage 1: masked score keys (streaming, HBM-bound) ----------------
// obj logits for anchor i are exactly ((float2*)obj)[i]  (cell*9 + k == i)
__global__ void k_keys(const float2* __restrict__ obj2, unsigned* __restrict__ keys, int n) {
    int i = blockIdx.x * blockDim.x + threadIdx.x;
    if (i >= n) return;
    float2 l = obj2[i];
    float score = 1.0f / (1.0f + __expf(l.x - l.y));   // softmax[1]
    float m = (score > THRESH) ? score : -1.0f;
    keys[i] = f2key(m);
}

// ---------------- stage 2: 3x8-bit radix select ----------------
// ctl[0]=prefix  ctl[1]=mask  ctl[2]=rank-remaining  ctl[3]=cand-count
__global__ void k_init(unsigned* hist, unsigned* ctl) {
    int t = threadIdx.x;
    hist[t] = 0;
    if (t == 0) { ctl[0] = 0u; ctl[1] = 0u; ctl[2] = TOPK; ctl[3] = 0u; }
}

__global__ void k_hist(const unsigned* __restrict__ keys, unsigned* __restrict__ hist,
                       const unsigned* __restrict__ ctl, int n, int shift) {
    __shared__ unsigned lh[256];
    lh[threadIdx.x] = 0;
    __syncthreads();
    unsigned prefix = ctl[0];
    unsigned mask   = ctl[1];
    int stride = gridDim.x * blockDim.x;
    for (int i = blockIdx.x * blockDim.x + threadIdx.x; i < n; i += stride) {
        unsigned key = keys[i];
        if ((key & mask) == prefix)
            atomicAdd(&lh[(key >> shift) & 0xFFu], 1u);
    }
    __syncthreads();
    unsigned c = lh[threadIdx.x];
    if (c) atomicAdd(&hist[threadIdx.x], c);
}

__global__ void k_select(unsigned* hist, unsigned* ctl, int shift) {
    if (threadIdx.x == 0) {
        unsigned r = ctl[2];
        unsigned cum = 0, b = 0, nr = r;
        for (int bin = 255; bin >= 0; --bin) {
            unsigned c = hist[bin];
            if (cum + c >= r) { b = (unsigned)bin; nr = r - cum; break; }
            cum += c;
        }
        ctl[0] |= (b << shift);
        ctl[1] = 0xFFFFFFFFu << shift;   // bits >= shift now fixed
        ctl[2] = nr;
    }
    __syncthreads();
    hist[threadIdx.x] = 0;               // ready for next pass
}

__global__ void k_compact(const unsigned* __restrict__ keys, unsigned* ctl,
                          unsigned long long* __restrict__ cand, int n) {
    unsigned prefix = ctl[0];
    unsigned mask   = ctl[1];            // 0xFFFFFF00 after 3 passes
    int stride = gridDim.x * blockDim.x;
    for (int i = blockIdx.x * blockDim.x + threadIdx.x; i < n; i += stride) {
        unsigned key = keys[i];
        if ((key & mask) >= prefix) {
            unsigned pos = atomicAdd(&ctl[3], 1u);
            if (pos < CANDMAX)
                cand[pos] = ((unsigned long long)key << 32) | (unsigned)(~(unsigned)i);
        }
    }
}

// ---------------- stage 3: bitonic sort of <=4096 candidates in LDS ----------------
__global__ void k_sort(unsigned long long* __restrict__ cand, const unsigned* __restrict__ ctl) {
    __shared__ unsigned long long v[CANDMAX];   // 32 KB of 320 KB WGP LDS
    int t = threadIdx.x;                         // 1024 threads
    unsigned cnt = ctl[3];
    if (cnt > CANDMAX) cnt = CANDMAX;
    const unsigned long long PAD = ((unsigned long long)f2key(-1.0f) << 32);
    for (int i = t; i < CANDMAX; i += 1024) v[i] = (i < (int)cnt) ? cand[i] : PAD;
    __syncthreads();
    for (int k = 2; k <= CANDMAX; k <<= 1) {
        for (int j = k >> 1; j > 0; j >>= 1) {
            for (int i = t; i < CANDMAX; i += 1024) {
                int ixj = i ^ j;
                if (ixj > i) {
                    unsigned long long a = v[i], b = v[ixj];
                    bool up = ((i & k) == 0);
                    bool sw = up ? (a < b) : (a > b);   // descending overall
                    if (sw) { v[i] = b; v[ixj] = a; }
                }
            }
            __syncthreads();
        }
    }
    for (int i = t; i < CANDMAX; i += 1024) cand[i] = v[i];
}

// ---------------- stage 4: decode only the 4000 winners ----------------
__global__ void k_gather(const unsigned long long* __restrict__ cand,
                         const float2* __restrict__ obj2, const float4* __restrict__ reg4,
                         const int* __restrict__ anchor_px,
                         float* __restrict__ out_b, float* __restrict__ out_o,
                         float4* __restrict__ aux, float* __restrict__ areaArr,
                         unsigned* __restrict__ valid) {
    int t = blockIdx.x * blockDim.x + threadIdx.x;
    if (t >= TOPK) return;
    unsigned long long v = cand[t];
    unsigned key = (unsigned)(v >> 32);
    unsigned idx = ~(unsigned)v;
    float ms = key2f(key);
    bool inrange = idx < (unsigned)N_ANCH;
    unsigned s = inrange ? idx : 0u;

    int kk = (int)(s % K_ANCH);
    int cell = (int)(s / K_ANCH);
    int x = cell % W_FEAT, y = cell / W_FEAT;
    float2 l  = obj2[s];                       // coalesced b64
    float4 rg = reg4[s];                       // coalesced b128
    float p1 = 1.0f / (1.0f + __expf(l.x - l.y));
    float aw = (float)anchor_px[kk];
    float cx = (x + 0.5f) * FIELD + rg.x * aw;
    float cy = (y + 0.5f) * FIELD + rg.y * aw;
    float w = aw * __expf(rg.z);
    float h = aw * __expf(rg.w);
    float x1 = fminf(fmaxf(cx - 0.5f * w, 0.f), IMG_W);
    float y1 = fminf(fmaxf(cy - 0.5f * h, 0.f), IMG_H);
    float x2 = fminf(fmaxf(cx + 0.5f * w, 0.f), IMG_W);
    float y2 = fminf(fmaxf(cy + 0.5f * h, 0.f), IMG_H);
    float bw = x2 - x1, bh = y2 - y1;

    out_b[4 * t + 0] = x1;  out_b[4 * t + 1] = y1;
    out_b[4 * t + 2] = bw;  out_b[4 * t + 3] = bh;
    out_o[2 * t + 0] = 1.0f - p1;
    out_o[2 * t + 1] = p1;
    aux[t] = make_float4(x1, y1, x2, y2);
    areaArr[t] = bw * bh;
    valid[t] = (inrange && ms > THRESH) ? 1u : 0u;
}

// ---------------- stage 5: WMMA IoU + suppression bitmask ----------------
// One wave per 16x32 tile. area_i + area_j computed as rank-2 outer product on
// the matrix pipe: A row m = [area_row[m], 1, 0, 0]; B col n = [1, area_col[n], 0, 0]^T.
__global__ void k_iou(const float4* __restrict__ aux, const float* __restrict__ area,
                      unsigned* __restrict__ sup) {
    const int ROWT = TOPK / 16;   // 250
    const int COLT = TOPK / 32;   // 125
    int wid  = blockIdx.x * (blockDim.x >> 5) + (threadIdx.x >> 5);  // uniform per wave
    int lane = threadIdx.x & 31;
    if (wid >= ROWT * COLT) return;    // whole-wave exit: EXEC stays all-ones for WMMA
    int tr = wid / COLT, tc = wid % COLT;
    int ri = tr * 16, cj = tc * 32;
    int  lm = lane & 15;
    bool hi = lane >= 16;

    __builtin_prefetch(&sup[(ri)*WORDS + tc], 1, 0);

    float4 rb = aux[ri + lm];     // row boxes: lane m holds row m (dup in hi half)
    float  rar = area[ri + lm];
    float4 cb = aux[cj + lane];   // col boxes: lane n holds col n (32 cols)
    float  car = area[cj + lane];

    v2f a;                         // A 16x4: lanes 0-15 K=0,1 ; lanes 16-31 K=2,3 (zero)
    a.x = hi ? 0.0f : rar;
    a.y = hi ? 0.0f : 1.0f;

    unsigned words[16];
#pragma unroll
    for (int r = 0; r < 16; ++r) words[r] = 0u;

#pragma unroll
    for (int h = 0; h < 2; ++h) {
        int src = h * 16 + lm;
        float bar = __shfl(car, src, 32);
        v2f b;                     // B 4x16: row0 = ones, row1 = area_col
        b.x = hi ? 0.0f : 1.0f;
        b.y = hi ? 0.0f : bar;
        v8f c = {};
        v8f d = __builtin_amdgcn_wmma_f32_16x16x4_f32(false, a, false, b,
                                                      (short)0, c, false, false);
        float cx1 = __shfl(cb.x, src, 32);
        float cy1 = __shfl(cb.y, src, 32);
        float cx2 = __shfl(cb.z, src, 32);
        float cy2 = __shfl(cb.w, src, 32);
        int gj = cj + h * 16 + lm;
#pragma unroll
        for (int r = 0; r < 8; ++r) {
            int m = r + (hi ? 8 : 0);
            float rx1 = __shfl(rb.x, m, 32);
            float ry1 = __shfl(rb.y, m, 32);
            float rx2 = __shfl(rb.z, m, 32);
            float ry2 = __shfl(rb.w, m, 32);
            float iw = fmaxf(fminf(rx2, cx2) - fmaxf(rx1, cx1), 0.0f);
            float ih = fmaxf(fminf(ry2, cy2) - fmaxf(ry1, cy1), 0.0f);
            float inter = iw * ih;
            float iou = inter / (d[r] - inter + 1e-8f);   // d[r] = area_i + area_j via WMMA
            int gi = ri + m;
            bool pred = (iou > NMS_IOU) && (gj > gi);
            unsigned msk = (unsigned)__ballot(pred);
            // lane L of msk -> (row = r + (L<16?0:8), col = L&15) within tile
            words[r]     |= (msk & 0xFFFFu)  << (h * 16);
            words[r + 8] |= (msk >> 16)      << (h * 16);
        }
    }
    if (lane == 0) {
#pragma unroll
        for (int r = 0; r < 16; ++r)
            sup[(ri + r) * WORDS + tc] = words[r];
    }
}

// ---------------- stage 6: serial greedy NMS over 125-word bitmasks ----------------
// Latency-bound serial chain: double-buffer each suppression row into LDS with
// GLOBAL_LOAD_ASYNC_TO_LDS (ASYNCcnt path) so row i+1 streams in while row i is
// being merged. Async loads complete in order, so asynccnt<=1 ensures row i landed.
__global__ void k_nms(const unsigned* __restrict__ sup, const unsigned* __restrict__ valid,
                      float* __restrict__ keepOut) {
    __shared__ unsigned removed[WORDS];
    int t = threadIdx.x;   // 128 threads = 4 waves
    if (t < WORDS) removed[t] = 0u;
#if defined(HAVE_ASYNC_LDS)
    __shared__ unsigned rowbuf[2][WORDS];
    if (t < WORDS) {
        GlobI32* g = (GlobI32*)(uintptr_t)(sup + t);
        LdsI32*  l = (LdsI32*)(unsigned)(uintptr_t)&rowbuf[0][t];
        __builtin_amdgcn_global_load_async_to_lds_b32(g, l, 0, 0);
    }
    __syncthreads();
    for (int i = 0; i < TOPK; ++i) {
        int cur = i & 1;
        if (i + 1 < TOPK && t < WORDS) {
            GlobI32* g = (GlobI32*)(uintptr_t)(sup + (size_t)(i + 1) * WORDS + t);
            LdsI32*  l = (LdsI32*)(unsigned)(uintptr_t)&rowbuf[cur ^ 1][t];
            __builtin_amdgcn_global_load_async_to_lds_b32(g, l, 0, 0);
        }
        bool kept = (valid[i] != 0u) && !((removed[i >> 5] >> (i & 31)) & 1u);
        __syncthreads();
        if (i + 1 < TOPK) __builtin_amdgcn_s_wait_asynccnt(1);
        else              __builtin_amdgcn_s_wait_asynccnt(0);
        if (kept && t < WORDS) removed[t] |= rowbuf[cur][t];
        if (t == 0) keepOut[i] = kept ? 1.0f : 0.0f;
        __syncthreads();
    }
#else
    __syncthreads();
    for (int i = 0; i < TOPK; ++i) {
        if (t < WORDS) __builtin_prefetch(&sup[(size_t)(i + 1) * WORDS + t], 0, 0);
        bool kept = (valid[i] != 0u) && !((removed[i >> 5] >> (i & 31)) & 1u);
        __syncthreads();
        if (kept && t < WORDS) removed[t] |= sup[(size_t)i * WORDS + t];
        if (t == 0) keepOut[i] = kept ? 1.0f : 0.0f;
        __syncthreads();
    }
#endif
}

extern "C" void kernel_launch(void* const* d_in, const int* in_sizes, int n_in,
                              void* d_out, int out_size, void* d_ws, size_t ws_size,
                              hipStream_t stream) {
    const float2* obj2 = (const float2*)d_in[0];  // [H,W,2K] -> float2 per anchor
    const float4* reg4 = (const float4*)d_in[1];  // [H,W,4K] -> float4 per anchor
    // d_in[2] = anchors (recomputed analytically on device)
    const int* anchor_px = (const int*)d_in[3];
    float* out = (float*)d_out;      // b[4000*4] | o[4000*2] | keep[4000]

    // workspace carve-up (all 16B-aligned)
    char* ws = (char*)d_ws;
    size_t off = 0;
    unsigned* keys = (unsigned*)(ws + off);            off += (size_t)N_ANCH * 4;
    unsigned* hist = (unsigned*)(ws + off);            off += 256 * 4;
    unsigned* ctl  = (unsigned*)(ws + off);            off += 16 * 4;
    off = (off + 15) & ~(size_t)15;
    unsigned long long* cand = (unsigned long long*)(ws + off); off += (size_t)CANDMAX * 8;
    float4*   aux   = (float4*)(ws + off);             off += (size_t)TOPK * 16;
    float*    areaA = (float*)(ws + off);              off += (size_t)TOPK * 4;
    unsigned* valid = (unsigned*)(ws + off);           off += (size_t)TOPK * 4;
    unsigned* sup   = (unsigned*)(ws + off);           off += (size_t)TOPK * WORDS * 4;

    k_init<<<1, 256, 0, stream>>>(hist, ctl);
    k_keys<<<(N_ANCH + 255) / 256, 256, 0, stream>>>(obj2, keys, N_ANCH);

    for (int shift = 24; shift >= 8; shift -= 8) {
        k_hist<<<2048, 256, 0, stream>>>(keys, hist, ctl, N_ANCH, shift);
        k_select<<<1, 256, 0, stream>>>(hist, ctl, shift);
    }
    k_compact<<<2048, 256, 0, stream>>>(keys, ctl, cand, N_ANCH);
    k_sort<<<1, 1024, 0, stream>>>(cand, ctl);
    k_gather<<<(TOPK + 255) / 256, 256, 0, stream>>>(cand, obj2, reg4, anchor_px,
                                                     out, out + TOPK * 4,
                                                     aux, areaA, valid);
    {
        const int tiles = (TOPK / 16) * (TOPK / 32);   // 31250 waves
        const int wavesPerBlock = 8;                   // 256 threads
        k_iou<<<(tiles + wavesPerBlock - 1) / wavesPerBlock, 256, 0, stream>>>(aux, areaA, sup);
    }
    k_nms<<<1, 128, 0, stream>>>(sup, valid, out + TOPK * 4 + TOPK * 2);
}